// Model_59261958750925
// MI455X (gfx1250) — compile-verified
//
#include <hip/hip_runtime.h>

typedef __attribute__((ext_vector_type(2))) float v2f;
typedef __attribute__((ext_vector_type(8))) float v8f;

#define N_NODES 16384
#define DDIM    512
#define E_EXP   4
#define F_DIM   512
#define KPAD    516   // 16B-aligned float4 rows; 4m bank skew -> conflict-free b64 A reads

// ---------------- kernel 1: zero per-expert counters ----------------
__global__ void init_counters(int* __restrict__ cnt) {
    if (threadIdx.x < E_EXP) cnt[threadIdx.x] = 0;
}

// ---------------- kernel 2: gate + argmax + counting-sort scatter ----------------
// One wave32 per node. softmax is monotone -> argmax over raw logits.
__global__ void __launch_bounds__(256)
route_kernel(const int* __restrict__ nodes,
             const float* __restrict__ embed_table,
             const float* __restrict__ gate_weight,
             int* __restrict__ cnt,
             int* __restrict__ bucket) {
    __shared__ float Wg[E_EXP * DDIM];   // 8 KB
    for (int i = threadIdx.x; i < E_EXP * DDIM; i += blockDim.x)
        Wg[i] = gate_weight[i];
    __syncthreads();

    const int wave = threadIdx.x >> 5;
    const int lane = threadIdx.x & 31;
    const int n = blockIdx.x * 8 + wave;
    if (n >= N_NODES) return;

    const float* er = embed_table + (size_t)nodes[n] * DDIM;
    float a0 = 0.f, a1 = 0.f, a2 = 0.f, a3 = 0.f;
    #pragma unroll 4
    for (int k = lane; k < DDIM; k += 32) {
        float x = er[k];                       // coalesced 128B per step
        a0 += x * Wg[0 * DDIM + k];
        a1 += x * Wg[1 * DDIM + k];
        a2 += x * Wg[2 * DDIM + k];
        a3 += x * Wg[3 * DDIM + k];
    }
    #pragma unroll
    for (int off = 16; off >= 1; off >>= 1) {  // wave32 reduction
        a0 += __shfl_xor(a0, off, 32);
        a1 += __shfl_xor(a1, off, 32);
        a2 += __shfl_xor(a2, off, 32);
        a3 += __shfl_xor(a3, off, 32);
    }
    if (lane == 0) {
        int best = 0; float bv = a0;           // strict > keeps first-max (jnp.argmax)
        if (a1 > bv) { bv = a1; best = 1; }
        if (a2 > bv) { bv = a2; best = 2; }
        if (a3 > bv) { bv = a3; best = 3; }
        int pos = atomicAdd(&cnt[best], 1);
        bucket[best * N_NODES + pos] = n;
    }
}

// ---------------- kernel 3: per-expert tiled GEMM via V_WMMA_F32_16X16X4_F32 ----------------
// grid.x = E_EXP * (N/16) row-tiles (worst case), grid.y = F/128 col blocks.
// Block: 256 threads = 8 waves; wave w computes a 16x16 C tile at cols cb + 16w.
__global__ void __launch_bounds__(256, 2)
moe_gemm_kernel(const int* __restrict__ nodes,
                const float* __restrict__ embed_table,
                const float* __restrict__ expert_weights,
                const int* __restrict__ cnt,
                const int* __restrict__ bucket,
                float* __restrict__ out) {
    __shared__ float As[16 * KPAD];   // 33 KB A tile (16 nodes x 512, padded)
    __shared__ int   nid[16];

    const int e     = blockIdx.x >> 10;        // / (N/16)
    const int tile  = blockIdx.x & 1023;
    const int count = cnt[e];
    const int row0  = tile * 16;
    if (row0 >= count) return;                 // uniform exit for empty tiles

    const int t  = threadIdx.x;
    if (t < 16) {
        int r = row0 + t;
        nid[t] = (r < count) ? bucket[e * N_NODES + r] : -1;
    }
    __syncthreads();

    // Stage A tile: each thread loads 32 floats (8 x float4) of one row chunk.
    {
        const int r  = t >> 4;                 // 0..15
        const int c0 = (t & 15) * 32;          // 0..480
        float* dst = &As[r * KPAD + c0];       // 16B aligned (KPAD*4 % 16 == 0)
        const int id = nid[r];
        if (id >= 0) {
            const float* src = embed_table + (size_t)nodes[id] * DDIM + c0;
            #pragma unroll
            for (int j = 0; j < 8; ++j)
                *(float4*)(dst + j * 4) = *(const float4*)(src + j * 4);
        } else {
            #pragma unroll
            for (int j = 0; j < 8; ++j)
                *(float4*)(dst + j * 4) = make_float4(0.f, 0.f, 0.f, 0.f);
        }
    }
    __syncthreads();

    const int wave = t >> 5;
    const int lane = t & 31;
    const int hi   = lane >> 4;                // K-pair select (0: K0/K1, 1: K2/K3)
    const int ml   = lane & 15;                // A: M index, B: N index
    const int ncol = blockIdx.y * 128 + wave * 16 + ml;
    const float* __restrict__ W = expert_weights + (size_t)e * DDIM * F_DIM;

    // Preload epilogue routing state into registers (each lane only ever
    // stores rows m = i + 8*hi): avoids per-store ds_load + exec-mask dance.
    float* optr[8];
    bool   ovalid[8];
    #pragma unroll
    for (int i = 0; i < 8; ++i) {
        const int m = i + 8 * hi;
        ovalid[i] = (row0 + m < count);
        const int id = ovalid[i] ? nid[m] : 0;
        optr[i] = out + (size_t)id * F_DIM + ncol;
    }

    v8f c = {0.f, 0.f, 0.f, 0.f, 0.f, 0.f, 0.f, 0.f};
    const float* arow = &As[ml * KPAD + 2 * hi];

    #pragma unroll 4
    for (int k0 = 0; k0 < DDIM; k0 += 4) {
        // A fragment: 16x4 f32, lanes 0-15 = M, VGPR0/1 = K{0,1} (lo half) / K{2,3} (hi half)
        v2f a = *(const v2f*)(arow + k0);      // 8B-aligned ds_load_b64, bank-skewed
        // B fragment: 4x16 f32, lanes 0-15 = N, same K striping as A
        v2f b;
        b.x = W[(size_t)(k0 + 2 * hi)     * F_DIM + ncol];
        b.y = W[(size_t)(k0 + 2 * hi + 1) * F_DIM + ncol];
        c = __builtin_amdgcn_wmma_f32_16x16x4_f32(
                /*neg_a=*/false, a, /*neg_b=*/false, b,
                /*c_mod=*/(short)0, c, /*reuse_a=*/false, /*reuse_b=*/false);
    }

    // C layout: VGPR i -> (M=i, N=ml) for lanes 0-15, (M=i+8, N=ml) for lanes 16-31
    #pragma unroll
    for (int i = 0; i < 8; ++i) {
        if (ovalid[i])
            *optr[i] = c[i];
    }
}

// ---------------- launcher ----------------
extern "C" void kernel_launch(void* const* d_in, const int* in_sizes, int n_in,
                              void* d_out, int out_size, void* d_ws, size_t ws_size,
                              hipStream_t stream) {
    const int*   nodes          = (const int*)d_in[0];
    const float* embed_table    = (const float*)d_in[1];
    const float* gate_weight    = (const float*)d_in[2];
    const float* expert_weights = (const float*)d_in[3];
    float*       out            = (float*)d_out;

    int* cnt    = (int*)d_ws;          // 4 counters
    int* bucket = cnt + 16;            // E*N node ids (256 KB)

    init_counters<<<1, 32, 0, stream>>>(cnt);
    route_kernel<<<N_NODES / 8, 256, 0, stream>>>(nodes, embed_table, gate_weight,
                                                  cnt, bucket);
    dim3 grid(E_EXP * (N_NODES / 16), F_DIM / 128);
    moe_gemm_kernel<<<grid, 256, 0, stream>>>(nodes, embed_table, expert_weights,
                                              cnt, bucket, out);
}